// InterContactHead_17411797418529
// MI455X (gfx1250) — compile-verified
//
#include <hip/hip_runtime.h>
#include <hip/hip_bf16.h>

// InterContactHead for MI455X (gfx1250).
// out[n,m] = dot(s[n], Wi) + dot(s[m], Wj) + dot(z[n,m,:], Wz) + b
// Memory-bound on streaming z (512 MB fp32). K-reduction done with
// V_WMMA_F32_16X16X4_F32 (fp32-exact, 32 chained WMMAs per 16 outputs).

typedef float v2f __attribute__((ext_vector_type(2)));
typedef float v8f __attribute__((ext_vector_type(8)));

#define CS 384
#define CZ 128
#define NN 1024

// --- Kernel 1: si[n] = s[n]·Wi, sj[n] = s[n]·Wj  (tiny: 1024 x 384 GEMV x2) ---
__global__ void __launch_bounds__(256)
sproj_kernel(const float* __restrict__ s, const float* __restrict__ W,
             float* __restrict__ ws) {
    int n = blockIdx.x * blockDim.x + threadIdx.x;
    if (n >= NN) return;
    const float* srow = s + (size_t)n * CS;
    const float* Wi = W;        // W[0, 0:384]
    const float* Wj = W + CS;   // W[0, 384:768]
    float accI = 0.f, accJ = 0.f;
    #pragma unroll 8
    for (int d = 0; d < CS; ++d) {
        float sv = srow[d];
        accI = fmaf(sv, Wi[d], accI);
        accJ = fmaf(sv, Wj[d], accJ);
    }
    ws[n]      = accI;
    ws[NN + n] = accJ;
}

// --- Kernel 2: main z reduction via WMMA f32 16x16x4 ---
// One wave handles 16 consecutive pixels p = pbase..pbase+15 (same n, consecutive m).
// A (16x4 f32): lanes 0-15 -> (M=lane, K=0,1), lanes 16-31 -> (M=lane-16, K=2,3),
// so each lane loads a contiguous float2 of z channels.
// B (4x16): every column = Wz chunk -> all D columns identical.
__global__ void __launch_bounds__(256)
zhead_kernel(const float* __restrict__ z,
             const float* __restrict__ W,
             const float* __restrict__ bias,
             const float* __restrict__ ws,
             float* __restrict__ out) {
    const int lane  = threadIdx.x & 31;
    const int wave  = threadIdx.x >> 5;
    const int gwave = blockIdx.x * 8 + wave;     // 8 waves / 256-thread block
    const int pbase = gwave * 16;                // 16 pixels per wave
    const int n      = pbase >> 10;              // pbase multiple of 16; same n for tile
    const int mbase  = pbase & (NN - 1);
    const int half   = lane >> 4;                // 0 -> K {0,1}; 1 -> K {2,3}
    const int mrow   = lane & 15;                // M row owned by this lane

    const float* zp = z + (size_t)(pbase + mrow) * CZ + half * 2;
    const float* Wz = W + 2 * CS + half * 2;     // W[0, 768:896], cache-resident

    v8f c = {0.f, 0.f, 0.f, 0.f, 0.f, 0.f, 0.f, 0.f};

    #pragma unroll 4
    for (int kc = 0; kc < CZ / 4; ++kc) {
        v2f a  = *(const v2f*)(zp + kc * 4);     // z channels for this lane's (M,K) slots
        v2f bm = *(const v2f*)(Wz + kc * 4);     // Wz broadcast across all N columns
        // (neg_a, A, neg_b, B, c_mod, C, reuse_a, reuse_b)
        c = __builtin_amdgcn_wmma_f32_16x16x4_f32(false, a, false, bm,
                                                  (short)0, c, false, false);
    }

    // D columns are all identical: lane 0 has rows M=0..7 in c[0..7],
    // lane 16 has rows M=8..15 in c[0..7].
    const float si_n = ws[n];
    const float b0   = bias[0];
    if (mrow == 0) {
        const int    rbase = pbase + half * 8;
        const float* sj    = ws + NN + mbase + half * 8;
        #pragma unroll
        for (int r = 0; r < 8; ++r) {
            out[rbase + r] = c[r] + si_n + sj[r] + b0;
        }
    }
}

extern "C" void kernel_launch(void* const* d_in, const int* in_sizes, int n_in,
                              void* d_out, int out_size, void* d_ws, size_t ws_size,
                              hipStream_t stream) {
    (void)in_sizes; (void)n_in; (void)out_size; (void)ws_size;
    const float* s    = (const float*)d_in[0];   // (1,1024,384)
    const float* z    = (const float*)d_in[1];   // (1,1024,1024,128)
    const float* W    = (const float*)d_in[2];   // (1,896)
    const float* bias = (const float*)d_in[3];   // (1,)
    float* out = (float*)d_out;                  // (1,1024,1024,1)
    float* ws  = (float*)d_ws;                   // [0:1024)=si, [1024:2048)=sj

    // si/sj precompute: 1024 threads
    sproj_kernel<<<(NN + 255) / 256, 256, 0, stream>>>(s, W, ws);

    // main: 1024*1024 pixels / 16 per wave / 8 waves per block = 8192 blocks
    const int total_waves = (NN * NN) / 16;
    zhead_kernel<<<total_waves / 8, 256, 0, stream>>>(z, W, bias, ws, out);
}